// GraphSAGELayer_4209067950556
// MI455X (gfx1250) — compile-verified
//
#include <hip/hip_runtime.h>
#include <math.h>

typedef float v2f __attribute__((ext_vector_type(2)));
typedef float v8f __attribute__((ext_vector_type(8)));

#define D      128
#define KDIM   256
#define EPS_   1e-30f
#define LN_EPS 1e-5f

// Monotone order-preserving map float -> uint32 so a single native
// global_atomic_max_u32 implements float atomic-max (incl. -inf).
__device__ __forceinline__ unsigned int enc_f32(float f) {
  unsigned int u = __float_as_uint(f);
  return (u & 0x80000000u) ? ~u : (u | 0x80000000u);
}
__device__ __forceinline__ float dec_f32(unsigned int e) {
  unsigned int u = (e & 0x80000000u) ? (e ^ 0x80000000u) : ~e;
  return __uint_as_float(u);
}

// ---------------- init: maxenc = enc(-inf), sum = 0, deg = 0 ----------------
__global__ void init_ws(unsigned int* __restrict__ maxenc,
                        float* __restrict__ sumbuf,
                        int* __restrict__ deg, int N) {
  int i = blockIdx.x * blockDim.x + threadIdx.x;
  int total = N * D;
  if (i < total) {
    maxenc[i] = 0x007FFFFFu;   // enc(-inf)
    sumbuf[i] = 0.0f;
  }
  if (i < N) deg[i] = 0;
}

// ---------------- pass 1: segment max (one wave per edge) -------------------
__global__ __launch_bounds__(256)
void edge_max(const float* __restrict__ x, const int* __restrict__ esrc,
              const int* __restrict__ edst, unsigned int* __restrict__ maxenc,
              int* __restrict__ deg, int E) {
  int e = blockIdx.x * 8 + (threadIdx.x >> 5);
  int lane = threadIdx.x & 31;
  if (e >= E) return;
  int src = esrc[e];
  int dst = edst[e];
  const float4 v = *(const float4*)(x + (size_t)src * D + lane * 4);
  unsigned int* mp = maxenc + (size_t)dst * D + lane * 4;
  atomicMax(mp + 0, enc_f32(v.x));
  atomicMax(mp + 1, enc_f32(v.y));
  atomicMax(mp + 2, enc_f32(v.z));
  atomicMax(mp + 3, enc_f32(v.w));
  if (lane == 0) atomicAdd(deg + dst, 1);
}

// ---------------- pass 2: segment sum of exp(x - mx) ------------------------
__global__ __launch_bounds__(256)
void edge_sum(const float* __restrict__ x, const int* __restrict__ esrc,
              const int* __restrict__ edst, const unsigned int* __restrict__ maxenc,
              float* __restrict__ sumbuf, int E) {
  int e = blockIdx.x * 8 + (threadIdx.x >> 5);
  int lane = threadIdx.x & 31;
  if (e >= E) return;
  int src = esrc[e];
  int dst = edst[e];
  const float4 v = *(const float4*)(x + (size_t)src * D + lane * 4);
  const unsigned int* mp = maxenc + (size_t)dst * D + lane * 4;
  float* sp = sumbuf + (size_t)dst * D + lane * 4;
  atomicAdd(sp + 0, expf(v.x - dec_f32(mp[0])));
  atomicAdd(sp + 1, expf(v.y - dec_f32(mp[1])));
  atomicAdd(sp + 2, expf(v.z - dec_f32(mp[2])));
  atomicAdd(sp + 3, expf(v.w - dec_f32(mp[3])));
}

// ------- fused: logsumexp finalize + [x|agg]@W + bias + LN + ReLU -----------
// One wave (32 lanes) computes a 16-row x 128-col output tile with
// V_WMMA_F32_16X16X4_F32 (fp32 end-to-end).
__global__ __launch_bounds__(32)
void gemm_ln(const float* __restrict__ x, const unsigned int* __restrict__ maxenc,
             const float* __restrict__ sumbuf, const int* __restrict__ deg,
             const float* __restrict__ W, const float* __restrict__ bias,
             const float* __restrict__ gamma, const float* __restrict__ beta,
             float* __restrict__ out, int N) {
  __shared__ float At[16][KDIM + 4];   // +4 pad -> conflict-free column reads
  __shared__ float Ht[16][D + 4];

  const int lane = threadIdx.x;
  const int m0 = blockIdx.x * 16;

  // Build A tile: cols [0,128) = x, cols [128,256) = finalized logsumexp agg.
  for (int idx = lane; idx < 16 * (KDIM / 4); idx += 32) {
    int r  = idx / (KDIM / 4);
    int c4 = (idx % (KDIM / 4)) * 4;
    int node = m0 + r;
    if (node >= N) node = N - 1;
    float4 v;
    if (c4 < D) {
      v = *(const float4*)(x + (size_t)node * D + c4);
    } else {
      int cc = c4 - D;
      bool has = deg[node] > 0;
      const unsigned int* mp = maxenc + (size_t)node * D + cc;
      const float4 s = *(const float4*)(sumbuf + (size_t)node * D + cc);
      v.x = has ? dec_f32(mp[0]) + logf(fmaxf(s.x, EPS_)) : 0.0f;
      v.y = has ? dec_f32(mp[1]) + logf(fmaxf(s.y, EPS_)) : 0.0f;
      v.z = has ? dec_f32(mp[2]) + logf(fmaxf(s.z, EPS_)) : 0.0f;
      v.w = has ? dec_f32(mp[3]) + logf(fmaxf(s.w, EPS_)) : 0.0f;
    }
    *(float4*)&At[r][c4] = v;
  }
  __syncthreads();

  v8f C[8] = {};                        // 8 n-tiles of 16x16 fp32 accum

  const int m     = lane & 15;          // row (A) / col-in-tile (B,D)
  const int khalf = (lane >> 4) << 1;   // lanes 16-31 hold K = k+2, k+3

  #pragma unroll 1
  for (int k = 0; k < KDIM; k += 4) {
    const int krow = k + khalf;
    const v2f a = *(const v2f*)&At[m][krow];        // A frag (2 VGPRs)
    const float* w0 = W + (size_t)krow * D + m;
    const float* w1 = w0 + D;
    #pragma unroll
    for (int t = 0; t < 8; ++t) {
      v2f bf;
      bf.x = w0[t * 16];
      bf.y = w1[t * 16];
      C[t] = __builtin_amdgcn_wmma_f32_16x16x4_f32(
          false, a, false, bf, (short)0, C[t], false, false);
    }
  }

  // Scatter D tiles + bias into LDS. C layout: lane<16 -> M=j, lane>=16 -> M=j+8.
  const int mbase = (lane >> 4) * 8;
  #pragma unroll
  for (int t = 0; t < 8; ++t) {
    const int n = t * 16 + m;
    const float bb = bias[n];
    #pragma unroll
    for (int j = 0; j < 8; ++j)
      Ht[mbase + j][n] = C[t][j] + bb;
  }
  __syncthreads();

  // LayerNorm + ReLU: lanes 0-15 each own one row of 128.
  if (lane < 16) {
    const int node = m0 + lane;
    if (node < N) {
      float s = 0.0f, s2 = 0.0f;
      #pragma unroll 4
      for (int c = 0; c < D; ++c) {
        float v = Ht[lane][c];
        s += v; s2 += v * v;
      }
      const float mean = s * (1.0f / D);
      const float var  = s2 * (1.0f / D) - mean * mean;
      const float rstd = rsqrtf(var + LN_EPS);
      float* op = out + (size_t)node * D;
      #pragma unroll 4
      for (int c = 0; c < D; ++c) {
        float v = (Ht[lane][c] - mean) * rstd * gamma[c] + beta[c];
        op[c] = fmaxf(v, 0.0f);
      }
    }
  }
}

extern "C" void kernel_launch(void* const* d_in, const int* in_sizes, int n_in,
                              void* d_out, int out_size, void* d_ws, size_t ws_size,
                              hipStream_t stream) {
  const float* x     = (const float*)d_in[0];
  const int*   esrc  = (const int*)d_in[1];
  const int*   edst  = (const int*)d_in[2];
  const float* W     = (const float*)d_in[3];
  const float* b     = (const float*)d_in[4];
  const float* gamma = (const float*)d_in[5];
  const float* beta  = (const float*)d_in[6];
  float* out = (float*)d_out;

  const int N = in_sizes[0] / D;
  const int E = in_sizes[1];

  // Workspace: maxenc[N*D] u32 | sumbuf[N*D] f32 | deg[N] i32  (~51.4 MB)
  unsigned int* maxenc = (unsigned int*)d_ws;
  float*        sumbuf = (float*)(maxenc + (size_t)N * D);
  int*          deg    = (int*)(sumbuf + (size_t)N * D);

  const int totalND = N * D;
  init_ws<<<(totalND + 255) / 256, 256, 0, stream>>>(maxenc, sumbuf, deg, N);
  edge_max<<<(E + 7) / 8, 256, 0, stream>>>(x, esrc, edst, maxenc, deg, E);
  edge_sum<<<(E + 7) / 8, 256, 0, stream>>>(x, esrc, edst, maxenc, sumbuf, E);
  gemm_ln<<<(N + 15) / 16, 32, 0, stream>>>(x, maxenc, sumbuf, deg, W, b, gamma,
                                            beta, out, N);
}